// BrainTemporalGNN_35897336660385
// MI455X (gfx1250) — compile-verified
//
#include <hip/hip_runtime.h>

#define NN 16384
#define HH 256
#define EB 524288
#define ET 262144
#define TA 107
#define NH (NN * HH)

typedef __attribute__((ext_vector_type(2))) float v2f;
typedef __attribute__((ext_vector_type(8))) float v8f;

__device__ __forceinline__ float wave_sum32(float v) {
#pragma unroll
  for (int off = 16; off > 0; off >>= 1) v += __shfl_xor(v, off, 32);
  return v;
}

__device__ __forceinline__ void atomadd(float* p, float v) {
  unsafeAtomicAdd(p, v);  // hardware global_atomic_add_f32
}

// ---------------------------------------------------------------- utilities
__global__ void k_fill(float* __restrict__ p, float v, int n) {
  int i = blockIdx.x * blockDim.x + threadIdx.x;
  if (i < n) p[i] = v;
}

__global__ void k_deg_accum(const int* __restrict__ col, const float* __restrict__ w,
                            float* __restrict__ deg, int E) {
  int i = blockIdx.x * blockDim.x + threadIdx.x;
  if (i < E) atomadd(&deg[col[i]], w[i]);
}

__global__ void k_rsqrt_inplace(float* __restrict__ p, int n) {
  int i = blockIdx.x * blockDim.x + threadIdx.x;
  if (i < n) {
    float d = p[i];
    p[i] = (d > 0.f) ? rsqrtf(d) : 0.f;
  }
}

__global__ void k_edge_norm(const int* __restrict__ row, const int* __restrict__ col,
                            const float* __restrict__ w, const float* __restrict__ dis,
                            float* __restrict__ nrm, int E) {
  int i = blockIdx.x * blockDim.x + threadIdx.x;
  if (i < E) nrm[i] = dis[row[i]] * w[i] * dis[col[i]];
}

// --------------------------------------------------- dense GEMM  Y = X @ W
// X:[nrows,256] row-major, W:[256,256] row-major, Y:[nrows,256].
// One 512-thread block (16 waves) per 16-row M-strip: X strip staged in LDS
// once (coalesced), each wave computes one 16x16 tile (nt = wave) with
// V_WMMA_F32_16X16X4_F32. LDS row stride 260 (== 4 mod 64 dwords) makes the
// wave's 64 dword reads hit all 64 banks exactly once (conflict-free) and
// keeps 8-byte alignment for float2 (ds_load_b64) reads.
__global__ __launch_bounds__(512) void k_gemm_256(const float* __restrict__ X,
                                                  const float* __restrict__ W,
                                                  float* __restrict__ Y) {
  __shared__ float xs[16][260];
  const int wave = threadIdx.x >> 5;
  const int lane = threadIdx.x & 31;
  const int hlf = lane >> 4;  // 0: K pair {0,1}, 1: K pair {2,3}
  const int l16 = lane & 15;
  const int m0 = blockIdx.x << 4;
  const int n0 = wave << 4;

  // stage 16x256 X strip, fully coalesced
#pragma unroll
  for (int f = threadIdx.x; f < 16 * 256; f += 512) {
    const int r = f >> 8, c = f & 255;
    xs[r][c] = X[(size_t)(m0 + r) * HH + c];
  }
  __syncthreads();

  const float* __restrict__ xr = &xs[l16][0];
  const float* __restrict__ wc = W + n0 + l16;  // B column for this lane
  v8f acc = {0.f, 0.f, 0.f, 0.f, 0.f, 0.f, 0.f, 0.f};
#pragma unroll 4
  for (int kk = 0; kk < HH; kk += 4) {
    const int ka = kk + (hlf << 1);
    v2f a, b;
    a.x = xr[ka];
    a.y = xr[ka + 1];
    b.x = wc[(size_t)ka * HH];
    b.y = wc[(size_t)(ka + 1) * HH];
    acc = __builtin_amdgcn_wmma_f32_16x16x4_f32(false, a, false, b, (short)0, acc,
                                                false, false);
  }
  // D layout: lane<16 -> rows m0..m0+7, lane>=16 -> rows m0+8..m0+15
  float* __restrict__ yr = Y + (size_t)(m0 + (hlf << 3)) * HH + n0 + l16;
#pragma unroll
  for (int j = 0; j < 8; ++j) yr[(size_t)j * HH] = acc[j];
}

// ------------------------------------------ fused temporal edge-weight MLP
// per edge: w = relu(LN(attr @ W1 + b1)) . w2 + b2   (never materializes hidden)
// block = 512 threads (16 waves) handles 16 edges x 256 hidden cols.
// attr tile staged in LDS (stride 132 == 4 mod 64, zero-padded K tail).
__global__ __launch_bounds__(512) void k_edge_mlp(
    const float* __restrict__ A, const float* __restrict__ W1,
    const float* __restrict__ b1, const float* __restrict__ g,
    const float* __restrict__ beta, const float* __restrict__ w2,
    const float* __restrict__ b2, float* __restrict__ outw) {
  __shared__ float as[16][132];       // staged attr rows, zero-padded past K=107
  __shared__ float tile[16][HH + 1];  // GEMM output tile for LN phase
  const int wave = threadIdx.x >> 5;
  const int lane = threadIdx.x & 31;
  const int hlf = lane >> 4;
  const int l16 = lane & 15;
  const int e0 = blockIdx.x << 4;
  const int n0 = wave << 4;

  // stage attr[e0..e0+15][0..107) into LDS, zero the pad columns
  for (int f = threadIdx.x; f < 16 * 132; f += 512) {
    const int r = f / 132, c = f - r * 132;
    as[r][c] = (c < TA) ? A[(size_t)(e0 + r) * TA + c] : 0.f;
  }
  __syncthreads();

  const float* __restrict__ ar = &as[l16][0];
  v8f acc = {0.f, 0.f, 0.f, 0.f, 0.f, 0.f, 0.f, 0.f};
#pragma unroll 4
  for (int kk = 0; kk < 108; kk += 4) {  // K=107 padded to 108 (A pad is zero)
    const int ka = kk + (hlf << 1);
    v2f a, b;
    a.x = ar[ka];
    a.y = ar[ka + 1];
    b.x = (ka < TA) ? W1[(size_t)ka * HH + n0 + l16] : 0.f;
    b.y = (ka + 1 < TA) ? W1[(size_t)(ka + 1) * HH + n0 + l16] : 0.f;
    acc = __builtin_amdgcn_wmma_f32_16x16x4_f32(false, a, false, b, (short)0, acc,
                                                false, false);
  }
#pragma unroll
  for (int j = 0; j < 8; ++j) tile[(hlf << 3) + j][n0 + l16] = acc[j];
  __syncthreads();

  // wave w performs LN + relu + dot(w2) for edge row w
  const int row = wave;
  float vals[8];
  float s = 0.f, sq = 0.f;
#pragma unroll
  for (int j = 0; j < 8; ++j) {
    const int c = lane + (j << 5);
    const float v = tile[row][c] + b1[c];
    vals[j] = v;
    s += v;
    sq += v * v;
  }
  s = wave_sum32(s);
  sq = wave_sum32(sq);
  const float mean = s * (1.f / HH);
  const float var = sq * (1.f / HH) - mean * mean;
  const float inv = rsqrtf(var + 1e-5f);
  float dot = 0.f;
#pragma unroll
  for (int j = 0; j < 8; ++j) {
    const int c = lane + (j << 5);
    float hv = (vals[j] - mean) * inv * g[c] + beta[c];
    hv = fmaxf(hv, 0.f);
    dot += hv * w2[c];
  }
  dot = wave_sum32(dot);
  if (lane == 0) outw[e0 + row] = dot + b2[0];
}

// ------------------------------------------------- edge scatter (GCN aggr)
// acc[col[e]] += norm[e] * h[row[e]], 64 threads/edge, 4 channels/thread.
// Working set (h + acc = 32MB) is L2-resident; atomics run in L2 atomic units.
__global__ __launch_bounds__(256) void k_scatter(const float* __restrict__ Hm,
                                                 const int* __restrict__ row,
                                                 const int* __restrict__ col,
                                                 const float* __restrict__ nrm,
                                                 float* __restrict__ acc, int E) {
  const long long t = (long long)blockIdx.x * blockDim.x + threadIdx.x;
  const int e = (int)(t >> 6);
  const int c4 = ((int)t & 63) << 2;
  if (e >= E) return;
  const int r = row[e], c = col[e];
  const float nv = nrm[e];
  const float4 v = *(const float4*)(Hm + (size_t)r * HH + c4);
  float* dst = acc + (size_t)c * HH + c4;
  atomadd(dst + 0, nv * v.x);
  atomadd(dst + 1, nv * v.y);
  atomadd(dst + 2, nv * v.z);
  atomadd(dst + 3, nv * v.w);
}

// -------------------- fused epilogue: self-loop + bias + residual + LN (+relu)
// val = acc[row] + dis[row]^2*h[row] + bias + res[row]; dst = LN(val)*g+b (relu?)
__global__ __launch_bounds__(256) void k_ln_epilogue(
    const float* __restrict__ acc, const float* __restrict__ hself,
    const float* __restrict__ dis, const float* __restrict__ bias,
    const float* __restrict__ res, const float* __restrict__ g,
    const float* __restrict__ b, float* __restrict__ dst, int do_relu) {
  const int wave = threadIdx.x >> 5;
  const int lane = threadIdx.x & 31;
  const int row = (blockIdx.x << 3) + wave;
  const float* ar = acc + (size_t)row * HH;
  const float sw = (dis != nullptr) ? dis[row] * dis[row] : 0.f;
  float vals[8];
  float s = 0.f, sq = 0.f;
#pragma unroll
  for (int j = 0; j < 8; ++j) {
    const int c = lane + (j << 5);
    float v = ar[c];
    if (hself) v += sw * hself[(size_t)row * HH + c];
    if (bias) v += bias[c];
    if (res) v += res[(size_t)row * HH + c];
    vals[j] = v;
    s += v;
    sq += v * v;
  }
  s = wave_sum32(s);
  sq = wave_sum32(sq);
  const float mean = s * (1.f / HH);
  const float var = sq * (1.f / HH) - mean * mean;
  const float inv = rsqrtf(var + 1e-5f);
#pragma unroll
  for (int j = 0; j < 8; ++j) {
    const int c = lane + (j << 5);
    float v = (vals[j] - mean) * inv * g[c] + b[c];
    if (do_relu) v = fmaxf(v, 0.f);
    dst[(size_t)row * HH + c] = v;
  }
}

extern "C" void kernel_launch(void* const* d_in, const int* in_sizes, int n_in,
                              void* d_out, int out_size, void* d_ws, size_t ws_size,
                              hipStream_t stream) {
  (void)in_sizes; (void)n_in; (void)out_size; (void)ws_size;
  const float* x   = (const float*)d_in[0];
  const int*   bei = (const int*)d_in[1];
  const float* bew = (const float*)d_in[2];
  const int*   tei = (const int*)d_in[3];
  const float* tea = (const float*)d_in[4];
  const float* Wb  = (const float*)d_in[5];
  const float* bb  = (const float*)d_in[6];
  const float* Wt  = (const float*)d_in[7];
  const float* bt  = (const float*)d_in[8];
  const float* lsg = (const float*)d_in[9];
  const float* lsb = (const float*)d_in[10];
  const float* ltg = (const float*)d_in[11];
  const float* ltb = (const float*)d_in[12];
  const float* pw1 = (const float*)d_in[13];
  const float* pb1 = (const float*)d_in[14];
  const float* pg  = (const float*)d_in[15];
  const float* pbt = (const float*)d_in[16];
  const float* pw2 = (const float*)d_in[17];
  const float* pb2 = (const float*)d_in[18];
  float* out = (float*)d_out;

  float* h   = (float*)d_ws;  // [NH]
  float* acc = h + NH;        // [NH]
  float* s   = acc + NH;      // [NH]
  float* dis = s + NH;        // [NN]  (deg -> 1/sqrt(deg))
  float* tw  = dis + NN;      // [ET]  temporal edge weights
  float* nrm = tw + ET;       // [EB]  per-edge norm (reused for temporal)

  const int* brow = bei;
  const int* bcol = bei + EB;
  const int* trow = tei;
  const int* tcol = tei + ET;

  // ---------------- bold GCN: h = x @ W_bold; aggregate; LN+relu -> s
  k_gemm_256<<<NN / 16, 512, 0, stream>>>(x, Wb, h);
  k_fill<<<NN / 256, 256, 0, stream>>>(dis, 1.0f, NN);  // self-loop weight
  k_deg_accum<<<EB / 256, 256, 0, stream>>>(bcol, bew, dis, EB);
  k_rsqrt_inplace<<<NN / 256, 256, 0, stream>>>(dis, NN);
  k_edge_norm<<<EB / 256, 256, 0, stream>>>(brow, bcol, bew, dis, nrm, EB);
  k_fill<<<NH / 256, 256, 0, stream>>>(acc, 0.0f, NH);
  k_scatter<<<(EB * 64) / 256, 256, 0, stream>>>(h, brow, bcol, nrm, acc, EB);
  k_ln_epilogue<<<NN / 8, 256, 0, stream>>>(acc, h, dis, bb, x, lsg, lsb, s, 1);

  // ---------------- temporal edge weights via fused MLP
  k_edge_mlp<<<ET / 16, 512, 0, stream>>>(tea, pw1, pb1, pg, pbt, pw2, pb2, tw);

  // ---------------- temporal GCN: h = s @ W_temp; aggregate; LN+relu -> acc
  k_gemm_256<<<NN / 16, 512, 0, stream>>>(s, Wt, h);
  k_fill<<<NN / 256, 256, 0, stream>>>(dis, 1.0f, NN);
  k_deg_accum<<<ET / 256, 256, 0, stream>>>(tcol, tw, dis, ET);
  k_rsqrt_inplace<<<NN / 256, 256, 0, stream>>>(dis, NN);
  k_edge_norm<<<ET / 256, 256, 0, stream>>>(trow, tcol, tw, dis, nrm, ET);
  k_fill<<<NH / 256, 256, 0, stream>>>(acc, 0.0f, NH);
  k_scatter<<<(ET * 64) / 256, 256, 0, stream>>>(h, trow, tcol, nrm, acc, ET);
  k_ln_epilogue<<<NN / 8, 256, 0, stream>>>(acc, h, dis, bt, s, ltg, ltb, acc, 1);

  // ---------------- final residual to x through struct LN -> out
  k_ln_epilogue<<<NN / 8, 256, 0, stream>>>(acc, nullptr, nullptr, nullptr, x, lsg,
                                            lsb, out, 0);
}